// ColmapReproj_87007447482946
// MI455X (gfx1250) — compile-verified
//
#include <hip/hip_runtime.h>

// COLMAP-style reprojection residuals on gfx1250 (MI455X).
// Memory-bound streaming kernel: 24 B/obs * 8M obs = 192 MB @ 23.3 TB/s ~ 8 us
// floor; ~0.3 GFLOP total -> VALU irrelevant, no matmul structure -> WMMA
// inapplicable. CDNA5 features used: 96 KB LDS camera table (>64 KB, legal on
// the WGP's 320 KB LDS; 3 blocks/WGP), ds_load_b128 camera gathers,
// non-temporal streaming loads/stores, global_prefetch_b8.
// ILP: 4 observations per lane per iteration for deep MLP.

typedef __attribute__((ext_vector_type(2))) float fv2;
typedef __attribute__((ext_vector_type(4))) float fv4;
typedef __attribute__((ext_vector_type(4))) int   iv4;

#define NCAM_MAX 2000
#define BLOCK    256
#define GRID     768

// Camera row layout in LDS (3 x float4 = 48B per camera):
//   [0] = (t.x, t.y, t.z, f)
//   [1] = (q.x, q.y, q.z, q.w)   (pre-normalized once, per camera)
//   [2] = (k1, k2, 0, 0)
__device__ __forceinline__ fv2 project_one(const float4* __restrict__ sc,
                                           const float* __restrict__ pts3,
                                           int c, int p, float mx, float my) {
    float4 a  = sc[3 * c + 0];
    float4 q  = sc[3 * c + 1];
    float4 kk = sc[3 * c + 2];
    const float* pp = pts3 + 3 * p;
    float px = pp[0], py = pp[1], pz = pp[2];
    // uv = qv x p
    float ux = q.y * pz - q.z * py;
    float uy = q.z * px - q.x * pz;
    float uz = q.x * py - q.y * px;
    // w = uv + qw * p
    float wx = fmaf(q.w, px, ux);
    float wy = fmaf(q.w, py, uy);
    float wz = fmaf(q.w, pz, uz);
    // c = qv x w
    float cx = q.y * wz - q.z * wy;
    float cy = q.z * wx - q.x * wz;
    float cz = q.x * wy - q.y * wx;
    // p_cam = p + 2*c + t
    float X = fmaf(2.0f, cx, px) + a.x;
    float Y = fmaf(2.0f, cy, py) + a.y;
    float Z = fmaf(2.0f, cz, pz) + a.z;
    float iz = 1.0f / Z;
    float u = X * iz;
    float v = Y * iz;
    float n = fmaf(u, u, v * v);
    float d = fmaf(n, fmaf(n, kk.y, kk.x), 1.0f);   // 1 + k1*n + k2*n^2
    float s = d * a.w;
    fv2 r;
    r.x = fmaf(u, s, -mx);
    r.y = fmaf(v, s, -my);
    return r;
}

__global__ __launch_bounds__(BLOCK) void ColmapReproj_kernel(
    const float* __restrict__ p2d,
    const int*   __restrict__ cidx,
    const int*   __restrict__ pidx,
    const float* __restrict__ pose,
    const float* __restrict__ pts3,
    float*       __restrict__ out,
    int M, int ncam) {
    // 96 KB static LDS: fits because gfx1250's WGP exposes 320 KB.
    __shared__ float4 s_cam[NCAM_MAX * 3];

    // ---- Stage camera table into LDS, folding quaternion normalization ----
    int nc = ncam < NCAM_MAX ? ncam : NCAM_MAX;
    for (int c = threadIdx.x; c < nc; c += BLOCK) {
        const float* pr = pose + 10 * c;
        float t0 = pr[0], t1 = pr[1], t2 = pr[2];
        float qx = pr[3], qy = pr[4], qz = pr[5], qw = pr[6];
        float f  = pr[7], k1 = pr[8], k2 = pr[9];
        float inv = 1.0f / sqrtf(qx * qx + qy * qy + qz * qz + qw * qw);
        qx *= inv; qy *= inv; qz *= inv; qw *= inv;
        s_cam[3 * c + 0] = make_float4(t0, t1, t2, f);
        s_cam[3 * c + 1] = make_float4(qx, qy, qz, qw);
        s_cam[3 * c + 2] = make_float4(k1, k2, 0.0f, 0.0f);
    }
    __syncthreads();

    // ---- Main streaming loop: 4 observations per lane per iteration ----
    const int nQuad  = M >> 2;
    const int stride = (int)(gridDim.x * blockDim.x);
    const iv4* ci4  = (const iv4*)cidx;
    const iv4* pi4  = (const iv4*)pidx;
    const fv4* m4   = (const fv4*)p2d;   // 2 obs per fv4
    fv4*       out4 = (fv4*)out;         // 2 obs per fv4

    for (int i = (int)(blockIdx.x * blockDim.x + threadIdx.x); i < nQuad; i += stride) {
        // Prefetch next grid-stride tile of the streaming arrays (speculative;
        // translation failures are silently dropped per ISA).
        __builtin_prefetch((const void*)(ci4 + i + stride), 0, 0);
        __builtin_prefetch((const void*)(pi4 + i + stride), 0, 0);
        __builtin_prefetch((const void*)(m4 + 2 * (i + stride)), 0, 0);

        iv4 cc = __builtin_nontemporal_load(ci4 + i);         // b128, TH_NT
        iv4 pp = __builtin_nontemporal_load(pi4 + i);         // b128, TH_NT
        fv4 m0 = __builtin_nontemporal_load(m4 + 2 * i);      // b128, TH_NT
        fv4 m1 = __builtin_nontemporal_load(m4 + 2 * i + 1);  // b128, TH_NT

        fv2 r0 = project_one(s_cam, pts3, cc.x, pp.x, m0.x, m0.y);
        fv2 r1 = project_one(s_cam, pts3, cc.y, pp.y, m0.z, m0.w);
        fv2 r2 = project_one(s_cam, pts3, cc.z, pp.z, m1.x, m1.y);
        fv2 r3 = project_one(s_cam, pts3, cc.w, pp.w, m1.z, m1.w);

        fv4 o0, o1;
        o0.x = r0.x; o0.y = r0.y; o0.z = r1.x; o0.w = r1.y;
        o1.x = r2.x; o1.y = r2.y; o1.z = r3.x; o1.w = r3.y;

        __builtin_nontemporal_store(o0, out4 + 2 * i);        // b128, TH_NT
        __builtin_nontemporal_store(o1, out4 + 2 * i + 1);    // b128, TH_NT
    }

    // ---- Tail (M % 4 != 0; M is a multiple of 4 in practice) ----
    if (blockIdx.x == 0) {
        for (int i = (nQuad << 2) + (int)threadIdx.x; i < M; i += BLOCK) {
            fv2 r = project_one(s_cam, pts3, cidx[i], pidx[i],
                                p2d[2 * i], p2d[2 * i + 1]);
            out[2 * i]     = r.x;
            out[2 * i + 1] = r.y;
        }
    }
}

extern "C" void kernel_launch(void* const* d_in, const int* in_sizes, int n_in,
                              void* d_out, int out_size, void* d_ws, size_t ws_size,
                              hipStream_t stream) {
    const float* p2d  = (const float*)d_in[0];  // (M, 2)
    const int*   cidx = (const int*)d_in[1];    // (M,)
    const int*   pidx = (const int*)d_in[2];    // (M,)
    const float* pose = (const float*)d_in[3];  // (NCAM, 10)
    const float* pts3 = (const float*)d_in[4];  // (NPTS, 3)
    float* out = (float*)d_out;                 // (M, 2)

    int M    = in_sizes[1];
    int ncam = in_sizes[3] / 10;

    int nQuad = M >> 2;
    int grid  = (nQuad + BLOCK - 1) / BLOCK;
    if (grid > GRID) grid = GRID;
    if (grid < 1)    grid = 1;

    ColmapReproj_kernel<<<grid, BLOCK, 0, stream>>>(p2d, cidx, pidx, pose, pts3,
                                                    out, M, ncam);
}